// AudioToTextCrossEntropyLoss_26929444946218
// MI455X (gfx1250) — compile-verified
//
#include <hip/hip_runtime.h>
#include <stdint.h>

typedef __attribute__((ext_vector_type(2))) float v2f;
typedef __attribute__((ext_vector_type(8))) float v8f;

// ---------------------------------------------------------------------------
// Kernel 1: one workgroup (256 threads) per row.
//   - async-copy the whole 128KB row into LDS (CDNA5 GLOBAL_LOAD_ASYNC_TO_LDS)
//   - pass 1 from LDS: row max
//   - pass 2 from LDS: Z = sum exp(x - max)
//   - window sum of raw logits over [t, t+p]
//   per_sample[b] = (max + log Z) - wsum/(p+1)
// ---------------------------------------------------------------------------
__global__ __launch_bounds__(256) void row_loss_kernel(
    const float* __restrict__ inputs,
    const long long* __restrict__ targets,
    const long long* __restrict__ plist,
    float* __restrict__ per_sample,
    int N)
{
    extern __shared__ __align__(16) float smem[];
    float* row = smem;          // N floats (row staged in LDS)
    float* red = smem + N;      // 256 floats reduction scratch

    const int tid = threadIdx.x;
    const int b   = blockIdx.x;
    const float* rowp = inputs + (size_t)b * (size_t)N;
    const int nvec = N >> 2;    // float4 chunks

    // ---- async global -> LDS copy of the entire row ----
    {
        unsigned long long gbase = (unsigned long long)(uintptr_t)rowp;
        uint32_t lds_base = (uint32_t)(uintptr_t)row;   // low 32 bits of flat LDS addr == LDS offset
        for (int c = tid; c < nvec; c += 256) {
            uint32_t byteoff = (uint32_t)c << 4;
            uint32_t ldsaddr = lds_base + byteoff;
            asm volatile("global_load_async_to_lds_b128 %0, %1, %2 offset:0"
                         :: "v"(ldsaddr), "v"(byteoff), "s"(gbase)
                         : "memory");
        }
        asm volatile("s_wait_asynccnt 0" ::: "memory");
    }
    __syncthreads();

    // ---- pass 1: row max (LDS, b128 reads) ----
    float m = -__builtin_inff();
    for (int c = tid; c < nvec; c += 256) {
        float4 v = *reinterpret_cast<const float4*>(row + ((size_t)c << 2));
        m = fmaxf(m, fmaxf(fmaxf(v.x, v.y), fmaxf(v.z, v.w)));
    }
    red[tid] = m;
    __syncthreads();
    for (int s = 128; s >= 1; s >>= 1) {
        if (tid < s) red[tid] = fmaxf(red[tid], red[tid + s]);
        __syncthreads();
    }
    const float rowmax = red[0];
    __syncthreads();

    // ---- pass 2: Z = sum exp(x - max) (LDS) ----
    float ssum = 0.0f;
    for (int c = tid; c < nvec; c += 256) {
        float4 v = *reinterpret_cast<const float4*>(row + ((size_t)c << 2));
        ssum += __expf(v.x - rowmax) + __expf(v.y - rowmax)
              + __expf(v.z - rowmax) + __expf(v.w - rowmax);
    }
    red[tid] = ssum;
    __syncthreads();
    for (int s = 128; s >= 1; s >>= 1) {
        if (tid < s) red[tid] += red[tid + s];
        __syncthreads();
    }
    const float Z = red[0];
    __syncthreads();

    // ---- window sum of raw logits over [t, t+p] (p < 64) ----
    const long long t = targets[b];
    const long long p = plist[b];
    float w = 0.0f;
    if (tid < 64 && (long long)tid <= p) w = row[t + tid];
    red[tid] = w;
    __syncthreads();
    for (int s = 128; s >= 1; s >>= 1) {
        if (tid < s) red[tid] += red[tid + s];
        __syncthreads();
    }

    if (tid == 0) {
        float count = (float)(p + 1);
        per_sample[b] = (rowmax + __logf(Z)) - red[0] / count;
    }
}

// ---------------------------------------------------------------------------
// Kernel 2: single wave32 reduces B per-row losses to the mean.
// Cross-lane sum done with V_WMMA_F32_16X16X4_F32:
//   A(16x4): VGPR0 = lane partials (K=0 rows for lanes 0-15, K=2 for 16-31),
//            VGPR1 = 0 (K=1,3).  B(4x16) = ones.  C = 0.
//   => D[m,n] = p[m] + p[m+16]  (independent of n)
//   Sum D's 8 VGPRs per lane, then lane0 + lane16 gives the full wave sum.
// ---------------------------------------------------------------------------
__global__ __launch_bounds__(32) void mean_reduce_kernel(
    const float* __restrict__ per_sample,
    float* __restrict__ out,
    int B)
{
    const int lane = threadIdx.x;
    float psum = 0.0f;
    for (int i = lane; i < B; i += 32) psum += per_sample[i];

#if __has_builtin(__builtin_amdgcn_wmma_f32_16x16x4_f32)
    v2f a;  a[0] = psum; a[1] = 0.0f;
    v2f bm; bm[0] = 1.0f; bm[1] = 1.0f;
    v8f c = {};
    v8f d = __builtin_amdgcn_wmma_f32_16x16x4_f32(
        /*neg_a=*/false, a, /*neg_b=*/false, bm,
        /*c_mod=*/(short)0, c, /*reuse_a=*/false, /*reuse_b=*/false);
    float t = d[0] + d[1] + d[2] + d[3] + d[4] + d[5] + d[6] + d[7];
    float total = __shfl(t, 0, 32) + __shfl(t, 16, 32);
#else
    float total = psum;
    for (int off = 16; off >= 1; off >>= 1)
        total += __shfl_xor(total, off, 32);
    total = __shfl(total, 0, 32);
#endif

    if (lane == 0) out[0] = total / (float)B;
}

extern "C" void kernel_launch(void* const* d_in, const int* in_sizes, int n_in,
                              void* d_out, int out_size, void* d_ws, size_t ws_size,
                              hipStream_t stream) {
    const float*     inputs  = (const float*)d_in[0];
    const long long* targets = (const long long*)d_in[1];
    const long long* plist   = (const long long*)d_in[2];
    float* out = (float*)d_out;
    float* ws  = (float*)d_ws;

    const int B = in_sizes[1];              // 1024
    const int N = in_sizes[0] / B;          // 32768

    const size_t shmem = (size_t)N * sizeof(float) + 256 * sizeof(float);

    row_loss_kernel<<<B, 256, shmem, stream>>>(inputs, targets, plist, ws, N);
    mean_reduce_kernel<<<1, 32, 0, stream>>>(ws, out, B);
}